// DSN_12498354831556
// MI455X (gfx1250) — compile-verified
//
#include <hip/hip_runtime.h>
#include <math.h>

// DSN few-shot classifier, fused for MI455X (gfx1250, wave32, WMMA).
//
// Math: dist[w][q] = ||q||^2 - ||Uk_w^T q||^2   (Uk orthonormal)
//       Uk_w = M_w V_w diag(1/sigma)  from 10x10 Gram eigenproblem.
// Heavy part = (16384x1024) @ (1024x45) f32 GEMM -> V_WMMA_F32_16X16X4_F32.
// HBM-bound: one 67MB pass over query (~2.9us at 23.3 TB/s); Uk (192KB) L2-resident.

typedef __attribute__((ext_vector_type(2))) float v2f;
typedef __attribute__((ext_vector_type(8))) float v8f;

#define N_WAY   5
#define N_SHOT  10
#define DIM     1024
#define N_QUERY 16384
#define NCOL    48              // 45 real columns (5 classes x 9), padded to 3 WMMA N-tiles
#define PAIR_STRIDE (NCOL * 2)  // packed layout: [k/2][n][2] so B fragment = one b64 load

// ---------------- kernel 1: Gram matrices G_w = M_w^T M_w (5 x 10 x 10) --------------
__global__ __launch_bounds__(512) void k_gram(const float* __restrict__ sup,
                                              float* __restrict__ G) {
    int t = threadIdx.x;
    if (t >= 500) return;
    int w = t / 100, rem = t % 100, i = rem / 10, j = rem % 10;
    // class w owns support rows {w, w+5, ..., w+45} (stable argsort of tiled labels)
    const float* a = sup + (size_t)(w + 5 * i) * DIM;
    const float* b = sup + (size_t)(w + 5 * j) * DIM;
    float s = 0.f;
    for (int k = 0; k < DIM; ++k) s = fmaf(a[k], b[k], s);
    G[t] = s;   // t == w*100 + i*10 + j
}

// ---------------- kernel 2: Jacobi eigensolver per class, emit W = V diag(1/sigma) ----
__global__ __launch_bounds__(32) void k_eigen(const float* __restrict__ G,
                                              float* __restrict__ W) {
    __shared__ float A[N_WAY][N_SHOT][N_SHOT];
    __shared__ float V[N_WAY][N_SHOT][N_SHOT];
    int w = threadIdx.x;
    if (w >= N_WAY) return;
    for (int i = 0; i < N_SHOT; ++i)
        for (int j = 0; j < N_SHOT; ++j) {
            A[w][i][j] = G[w * 100 + i * 10 + j];
            V[w][i][j] = (i == j) ? 1.f : 0.f;
        }
    for (int sweep = 0; sweep < 16; ++sweep) {
        for (int p = 0; p < N_SHOT - 1; ++p)
            for (int q = p + 1; q < N_SHOT; ++q) {
                float apq = A[w][p][q];
                if (fabsf(apq) < 1e-12f) continue;
                float app = A[w][p][p], aqq = A[w][q][q];
                float tau = (aqq - app) / (2.f * apq);
                float tt  = ((tau >= 0.f) ? 1.f : -1.f) / (fabsf(tau) + sqrtf(1.f + tau * tau));
                float c   = 1.f / sqrtf(1.f + tt * tt);
                float s   = tt * c;
                for (int r = 0; r < N_SHOT; ++r) {
                    float arp = A[w][r][p], arq = A[w][r][q];
                    A[w][r][p] = c * arp - s * arq;
                    A[w][r][q] = s * arp + c * arq;
                }
                for (int r = 0; r < N_SHOT; ++r) {
                    float apr = A[w][p][r], aqr = A[w][q][r];
                    A[w][p][r] = c * apr - s * aqr;
                    A[w][q][r] = s * apr + c * aqr;
                }
                for (int r = 0; r < N_SHOT; ++r) {
                    float vrp = V[w][r][p], vrq = V[w][r][q];
                    V[w][r][p] = c * vrp - s * vrq;
                    V[w][r][q] = s * vrp + c * vrq;
                }
            }
    }
    // drop the smallest eigenvalue direction (== keep top 9; projector is order-invariant)
    int jmin = 0; float lmin = A[w][0][0];
    for (int j = 1; j < N_SHOT; ++j) {
        float l = A[w][j][j];
        if (l < lmin) { lmin = l; jmin = j; }
    }
    int jj = 0;
    for (int j = 0; j < N_SHOT; ++j) {
        if (j == jmin) continue;
        float lam = fmaxf(A[w][j][j], 1e-12f);
        float inv = rsqrtf(lam);            // 1/sigma
        for (int i = 0; i < N_SHOT; ++i)
            W[(w * N_SHOT + i) * 9 + jj] = V[w][i][j] * inv;
        ++jj;
    }
}

// ---------------- kernel 3: pack Uk = M_w W_w into k-pair-interleaved [512][48][2] ----
__global__ __launch_bounds__(256) void k_pack(const float* __restrict__ sup,
                                              const float* __restrict__ W,
                                              float* __restrict__ P) {
    int idx = blockIdx.x * 256 + threadIdx.x;   // DIM*NCOL = 49152
    int k = idx / NCOL;
    int n = idx % NCOL;
    float v = 0.f;
    if (n < 45) {
        int w = n / 9, j = n % 9;
        for (int i = 0; i < N_SHOT; ++i)
            v = fmaf(sup[(size_t)(w + 5 * i) * DIM + k], W[(w * N_SHOT + i) * 9 + j], v);
    }
    P[(size_t)(k >> 1) * PAIR_STRIDE + n * 2 + (k & 1)] = v;
}

// ---------------- kernel 4: fused coeff-GEMM + dist + log_softmax ---------------------
// 1 wave = 16 queries; 256-thread block = 8 waves = 128 queries; 128 blocks.
__global__ __launch_bounds__(256) void k_main(const float* __restrict__ query,
                                              const float* __restrict__ P,
                                              float* __restrict__ out) {
    __shared__ float lds[8][NCOL * 16 + 32];    // per wave: coeff^2 [48][16] + qnorm[32]
    int lane = threadIdx.x & 31;
    int wave = threadIdx.x >> 5;
    int lo = lane & 15, hi = lane >> 4;
    int qbase = (blockIdx.x * 8 + wave) * 16;

    // A (16x4 f32): lanes 0-15 hold K={k,k+1} (v0,v1), lanes 16-31 K={k+2,k+3}
    const float* aptr = query + (size_t)(qbase + lo) * DIM + 2 * hi;
    // B (4x16 f32): mirrored layout; packed pairs make each fragment a single b64 load
    const float* bptr = P + hi * PAIR_STRIDE + lo * 2;

    v8f acc0 = {}, acc1 = {}, acc2 = {};
    float qn = 0.f;

#pragma unroll 4
    for (int k = 0; k < DIM; k += 4) {
        v2f a = *(const v2f*)(aptr + k);
        const float* bk = bptr + (size_t)(k >> 1) * PAIR_STRIDE;
        v2f b0 = *(const v2f*)(bk);        // N-tile 0: cols  0..15
        v2f b1 = *(const v2f*)(bk + 32);   // N-tile 1: cols 16..31
        v2f b2 = *(const v2f*)(bk + 64);   // N-tile 2: cols 32..47
        qn = fmaf(a.x, a.x, fmaf(a.y, a.y, qn));   // ||q||^2 partial (lanes m, m+16)
        acc0 = __builtin_amdgcn_wmma_f32_16x16x4_f32(false, a, false, b0, (short)0, acc0, false, false);
        acc1 = __builtin_amdgcn_wmma_f32_16x16x4_f32(false, a, false, b1, (short)0, acc1, false, false);
        acc2 = __builtin_amdgcn_wmma_f32_16x16x4_f32(false, a, false, b2, (short)0, acc2, false, false);
    }

    // C/D layout: VGPR r, lanes 0-15 -> (M=r, N=lane); lanes 16-31 -> (M=r+8, N=lane-16)
    float* L = lds[wave];
#pragma unroll
    for (int r = 0; r < 8; ++r) {
        int m = r + 8 * hi;
        L[(lo     ) * 16 + m] = acc0[r] * acc0[r];
        L[(lo + 16) * 16 + m] = acc1[r] * acc1[r];
        L[(lo + 32) * 16 + m] = acc2[r] * acc2[r];
    }
    L[NCOL * 16 + lane] = qn;
    __syncthreads();

    if (lane < 16) {
        int m = lane;
        float q2 = L[NCOL * 16 + m] + L[NCOL * 16 + 16 + m];
        float logit[N_WAY];
        float mx = -1e30f;
        for (int w = 0; w < N_WAY; ++w) {
            float s = 0.f;
            for (int j = 0; j < 9; ++j) s += L[(w * 9 + j) * 16 + m];
            logit[w] = s - q2;             // -dist
            mx = fmaxf(mx, logit[w]);
        }
        float se = 0.f;
        for (int w = 0; w < N_WAY; ++w) se += expf(logit[w] - mx);
        float lse = mx + logf(se);
        float* o = out + (size_t)(qbase + m) * N_WAY;
        for (int w = 0; w < N_WAY; ++w) o[w] = logit[w] - lse;
    }
}

extern "C" void kernel_launch(void* const* d_in, const int* in_sizes, int n_in,
                              void* d_out, int out_size, void* d_ws, size_t ws_size,
                              hipStream_t stream) {
    (void)in_sizes; (void)n_in; (void)out_size; (void)ws_size;
    const float* sup   = (const float*)d_in[0];   // train_imgs (50 x 1024) f32
    const float* query = (const float*)d_in[2];   // query_imgs (16384 x 1024) f32
    float* out = (float*)d_out;                   // (16384 x 5) f32
    float* ws  = (float*)d_ws;

    float* G = ws;            // 500 floats
    float* W = ws + 512;      // 450 floats
    float* P = ws + 1024;     // 512 * 96 = 49152 floats (~192 KB)

    k_gram <<<1, 512, 0, stream>>>(sup, G);
    k_eigen<<<1, 32, 0, stream>>>(G, W);
    k_pack <<<(DIM * NCOL) / 256, 256, 0, stream>>>(sup, W, P);
    k_main <<<N_QUERY / 128, 256, 0, stream>>>(query, P, out);
}